// TemporalAdversarialGNN_78606491451780
// MI455X (gfx1250) — compile-verified
//
#include <hip/hip_runtime.h>

// ---------------------------------------------------------------------------
// TemporalAdversarialGNN for MI455X (gfx1250, wave32)
//
//   agg  = segment_sum(x[src] -> (dst,t))          [N, T*C] = [N, 512]
//   h1   = relu(agg @ Wt1 + bt1)                   [N, 64]
//   agg2 = segment_sum(h1[src] -> (dst,t))         [N, 512]
//   h2   = relu(agg2 @ Wt2 + bt2)                  [N, 64]
//   hid  = relu(h2 @ Wa1 + ba1)                    [N, 128]
//   out  = hid @ Wa2 + ba2 + h1                    [N, 64]
//
// All GEMMs use V_WMMA_F32_16X16X4_F32 (fp32-exact vs the reference).
// ---------------------------------------------------------------------------

typedef float v2f __attribute__((ext_vector_type(2)));
typedef float v8f __attribute__((ext_vector_type(8)));

#define NNODES 50000
#define TSTEPS 8
#define CFEAT  64
#define NEDGES 800000

// ---------------------------------------------------------------- zero fill
__global__ __launch_bounds__(256) void tagnn_zero_kernel(float* __restrict__ p,
                                                         int nfloat4) {
  int i = blockIdx.x * blockDim.x + threadIdx.x;
  int stride = gridDim.x * blockDim.x;
  float4 z = make_float4(0.f, 0.f, 0.f, 0.f);
  for (; i < nfloat4; i += stride) ((float4*)p)[i] = z;
}

// ------------------------------------------------------- edge scatter (E,C)
// 64 threads per edge, one float channel each. agg layout:
//   agg[(dst*T + t)*C + c]  ==  cat[dst, t*C + c]  (node rows contiguous, 512)
__global__ __launch_bounds__(256) void tagnn_scatter_kernel(
    const float* __restrict__ feat,   // [N, C]
    const int* __restrict__ src,      // [E]
    const int* __restrict__ dst,      // [E]
    const int* __restrict__ tix,      // [E]
    float* __restrict__ agg) {        // [N*T, C]
  int e = blockIdx.x * 4 + (threadIdx.x >> 6);
  int c = threadIdx.x & 63;
  if (e >= NEDGES) return;
  int s = src[e];
  int d = dst[e];
  int t = tix[e];
  float v = feat[(size_t)s * CFEAT + c];
  float* p = agg + ((size_t)d * TSTEPS + t) * CFEAT + c;
  __hip_atomic_fetch_add(p, v, __ATOMIC_RELAXED, __HIP_MEMORY_SCOPE_AGENT);
}

// ------------------------------------------------- WMMA fp32 GEMM + epilogue
// out[m, n] = (relu?)(A[m, :K] @ W[:K, n] + bias[n]) (+ addend[m, n])
// One wave -> one 16x16 output tile. Block = 256 threads = 8 waves.
//   Cout = 64  -> 4 col tiles x 2 row tiles (32 rows / block)
//   Cout = 128 -> 8 col tiles x 1 row tile  (16 rows / block)
// A tile staged in LDS with coalesced float4 copies (block rows contiguous).
__global__ __launch_bounds__(256) void tagnn_wmma_gemm_kernel(
    const float* __restrict__ A,       // [nrows, K]
    const float* __restrict__ W,       // [K, Cout]
    const float* __restrict__ bias,    // [Cout]
    const float* __restrict__ addend,  // [nrows, Cout] or nullptr
    float* __restrict__ out,           // [nrows, Cout]
    int nrows, int K, int Cout, int applyRelu) {
  extern __shared__ float lds[];

  const int colTiles = Cout >> 4;        // 4 or 8
  const int rowTiles = 8 / colTiles;     // 2 or 1
  const int rowsPerBlock = rowTiles * 16;
  const int row0 = blockIdx.x * rowsPerBlock;

  // ---- cooperative A-tile load: one contiguous [rowsPerBlock*K] region ----
  const int totalFloats = rowsPerBlock * K;       // K is a multiple of 4
  const float* abase = A + (size_t)row0 * K;
  const long maxIdx = (long)nrows * K;
  for (int i = threadIdx.x * 4; i < totalFloats; i += blockDim.x * 4) {
    float4 v;
    if ((long)row0 * K + i + 3 < maxIdx) {
      v = *(const float4*)(abase + i);
    } else {
      v = make_float4(0.f, 0.f, 0.f, 0.f);   // rows past N contribute nothing
    }
    *(float4*)(lds + i) = v;
  }
  __syncthreads();

  const int lane = threadIdx.x & 31;
  const int wave = threadIdx.x >> 5;
  const int half = lane >> 4;   // selects K pair {0,1} vs {2,3} (A) / rows (B)
  const int lm   = lane & 15;   // A row within tile / B-D column within tile
  const int rowTile = wave / colTiles;
  const int colTile = wave % colTiles;
  const int col = colTile * 16 + lm;

  const float* aRow = lds + (size_t)(rowTile * 16 + lm) * K;

  v8f acc = {};
  for (int k0 = 0; k0 < K; k0 += 4) {
    const int ka = k0 + 2 * half;
    // A 16x4 f32 frag: lane(lm,half) holds A[m=lm][ka], A[m=lm][ka+1]
    v2f a;
    a.x = aRow[ka];
    a.y = aRow[ka + 1];
    // B 4x16 f32 frag: lane(lm,half) holds W[ka][col], W[ka+1][col]
    v2f b;
    b.x = W[(size_t)ka * Cout + col];
    b.y = W[(size_t)(ka + 1) * Cout + col];
    acc = __builtin_amdgcn_wmma_f32_16x16x4_f32(
        /*neg_a=*/false, a, /*neg_b=*/false, b,
        /*c_mod=*/(short)0, acc, /*reuse_a=*/false, /*reuse_b=*/false);
  }

  // ---- epilogue: D vgpr r -> (m = half*8 + r, n = lm) ----
  const float bv = bias[col];
#pragma unroll
  for (int r = 0; r < 8; ++r) {
    const int m = row0 + rowTile * 16 + half * 8 + r;
    if (m < nrows) {
      float v = acc[r] + bv;
      if (applyRelu) v = fmaxf(v, 0.f);
      if (addend) v += addend[(size_t)m * Cout + col];
      out[(size_t)m * Cout + col] = v;
    }
  }
}

// ---------------------------------------------------------------------------
extern "C" void kernel_launch(void* const* d_in, const int* in_sizes, int n_in,
                              void* d_out, int out_size, void* d_ws,
                              size_t ws_size, hipStream_t stream) {
  const float* x   = (const float*)d_in[0];
  const int*   ei  = (const int*)d_in[1];   // [2, E] row-major
  const int*   ti  = (const int*)d_in[2];   // [E]
  const float* Wt1 = (const float*)d_in[3];
  const float* bt1 = (const float*)d_in[4];
  const float* Wt2 = (const float*)d_in[5];
  const float* bt2 = (const float*)d_in[6];
  const float* Wa1 = (const float*)d_in[7];
  const float* ba1 = (const float*)d_in[8];
  const float* Wa2 = (const float*)d_in[9];
  const float* ba2 = (const float*)d_in[10];
  float* out = (float*)d_out;

  const int* src = ei;
  const int* dst = ei + NEDGES;

  const size_t aggN = (size_t)TSTEPS * NNODES * CFEAT;  // 25.6M floats
  float* agg = (float*)d_ws;
  float* h1  = agg + aggN;
  float* h2  = h1 + (size_t)NNODES * CFEAT;
  float* hid = agg;  // agg is dead once h2 exists; reuse for [N,128] hidden

  const int scatterGrid = (NEDGES + 3) / 4;
  const int zeroGrid = 2048;
  const int grid64  = (NNODES + 31) / 32;  // Cout=64 gemm: 32 rows/block
  const int grid128 = (NNODES + 15) / 16;  // Cout=128 gemm: 16 rows/block

  // ---- pass 1: agg = scatter(x); h1 = relu(agg @ Wt1 + bt1) ----
  tagnn_zero_kernel<<<zeroGrid, 256, 0, stream>>>(agg, (int)(aggN / 4));
  tagnn_scatter_kernel<<<scatterGrid, 256, 0, stream>>>(x, src, dst, ti, agg);
  tagnn_wmma_gemm_kernel<<<grid64, 256, 32 * 512 * sizeof(float), stream>>>(
      agg, Wt1, bt1, nullptr, h1, NNODES, TSTEPS * CFEAT, CFEAT, 1);

  // ---- pass 2: agg = scatter(h1); h2 = relu(agg @ Wt2 + bt2) ----
  tagnn_zero_kernel<<<zeroGrid, 256, 0, stream>>>(agg, (int)(aggN / 4));
  tagnn_scatter_kernel<<<scatterGrid, 256, 0, stream>>>(h1, src, dst, ti, agg);
  tagnn_wmma_gemm_kernel<<<grid64, 256, 32 * 512 * sizeof(float), stream>>>(
      agg, Wt2, bt2, nullptr, h2, NNODES, TSTEPS * CFEAT, CFEAT, 1);

  // ---- adversary MLP; final conv output is h1 + adv (same scatter as pass1)
  tagnn_wmma_gemm_kernel<<<grid128, 256, 16 * 64 * sizeof(float), stream>>>(
      h2, Wa1, ba1, nullptr, hid, NNODES, CFEAT, 128, 1);
  tagnn_wmma_gemm_kernel<<<grid64, 256, 32 * 128 * sizeof(float), stream>>>(
      hid, Wa2, ba2, /*addend=*/h1, out, NNODES, 128, CFEAT, 0);
}